// E2EBlock_24979529794351
// MI455X (gfx1250) — compile-verified
//
#include <hip/hip_runtime.h>

typedef __attribute__((ext_vector_type(2))) float v2f;
typedef __attribute__((ext_vector_type(8))) float v8f;

#define B_  64
#define C_  32
#define P_  64
#define D_  200
#define K_  (C_ * D_)     /* 6400 */
#define DD_ (D_ * D_)     /* 40000 */

// MODE 0: out[b,p,h] = sum_{c,w} x[b,c,h,w] * wgt[p,c,w]   (A rows contiguous)
// MODE 1: out[b,p,w] = sum_{c,h} x[b,c,h,w] * wgt[p,c,h]   (A rows stride-D)
// One wave (32 threads) per 16x16 output tile; K accumulated via
// v_wmma_f32_16x16x4_f32 with two accumulators to break the RAW chain.
template <int MODE>
__global__ __launch_bounds__(32) void gemm_wmma(const float* __restrict__ x,
                                                const float* __restrict__ wgt,
                                                float* __restrict__ outp) {
  const int lane = threadIdx.x;     // 0..31
  const int half = lane >> 4;       // 0 | 1
  const int m    = lane & 15;
  const int koff = half * 2;        // K sub-offset for this half-wave

  const int mt = blockIdx.x;        // 0..12  M tile (h for MODE0, w for MODE1)
  const int nt = blockIdx.y;        // 0..3   N tile (p)
  const int b  = blockIdx.z;        // 0..63

  const int m0 = mt * 16;
  const int p0 = nt * 16;

  const float* xb   = x + (size_t)b * C_ * DD_;
  const float* wrow = wgt + (size_t)(p0 + m) * K_;   // this lane's weight row

  int mc = m0 + m;                  // clamp ragged last tile for loads
  if (mc > D_ - 1) mc = D_ - 1;

  v8f acc0 = {};
  v8f acc1 = {};

  for (int c = 0; c < C_; ++c) {
    const float* xc = xb + (size_t)c * DD_;
    const float* wc = wrow + c * D_;
#pragma unroll 5
    for (int r = 0; r < D_; r += 8) {
      // ---- k-group at r -> acc0 ----
      {
        v2f a, bb;
        if (MODE == 0) {
          const float* pa = xc + (size_t)mc * D_ + (r + koff);
          a[0] = pa[0]; a[1] = pa[1];
        } else {
          const float* pa = xc + (size_t)(r + koff) * D_ + mc;
          a[0] = pa[0]; a[1] = pa[D_];
        }
        const float* pb = wc + r + koff;
        bb[0] = pb[0]; bb[1] = pb[1];
        acc0 = __builtin_amdgcn_wmma_f32_16x16x4_f32(false, a, false, bb,
                                                     (short)0, acc0, false, false);
      }
      // ---- k-group at r+4 -> acc1 ----
      {
        v2f a, bb;
        if (MODE == 0) {
          const float* pa = xc + (size_t)mc * D_ + (r + 4 + koff);
          a[0] = pa[0]; a[1] = pa[1];
        } else {
          const float* pa = xc + (size_t)(r + 4 + koff) * D_ + mc;
          a[0] = pa[0]; a[1] = pa[D_];
        }
        const float* pb = wc + r + 4 + koff;
        bb[0] = pb[0]; bb[1] = pb[1];
        acc1 = __builtin_amdgcn_wmma_f32_16x16x4_f32(false, a, false, bb,
                                                     (short)0, acc1, false, false);
      }
    }
  }

  v8f acc = acc0 + acc1;

  // C/D layout: VGPR r holds M = r + 8*half, lane&15 = N. Rows mrow..mrow+7
  // are contiguous in the [B,P,D] output -> two float4 stores per lane.
  const int mrow = m0 + half * 8;
  if (mrow < D_) {    // 200 % 8 == 0 -> all-or-nothing per 8-row group
    float* po = outp + ((size_t)b * P_ + p0 + m) * D_ + mrow;
    *(float4*)po       = make_float4(acc[0], acc[1], acc[2], acc[3]);
    *(float4*)(po + 4) = make_float4(acc[4], acc[5], acc[6], acc[7]);
  }
}

// out[b,p,h,w] = A[b,p,h] + Bm[b,p,w] ; pure-bandwidth pass, float4 stores.
// blockIdx.y = b*P+p (4096), blockIdx.x covers D*D/4 float4's of one image.
__global__ __launch_bounds__(256) void outer_sum(const float* __restrict__ A,
                                                 const float* __restrict__ Bm,
                                                 float* __restrict__ out) {
  const int t = blockIdx.x * blockDim.x + threadIdx.x;
  const int e = t * 4;                      // element offset inside one image
  if (e >= DD_) return;
  const int h = e / D_;
  const int w = e - h * D_;                 // multiple of 4 (D_%4==0)
  const size_t bp = blockIdx.y;

  const float av = A[bp * D_ + h];
  const float* bm = Bm + bp * D_ + w;
  float4 o;
  o.x = av + bm[0];
  o.y = av + bm[1];
  o.z = av + bm[2];
  o.w = av + bm[3];
  *(float4*)(out + bp * (size_t)DD_ + e) = o;
}

extern "C" void kernel_launch(void* const* d_in, const int* in_sizes, int n_in,
                              void* d_out, int out_size, void* d_ws, size_t ws_size,
                              hipStream_t stream) {
  (void)in_sizes; (void)n_in; (void)out_size; (void)ws_size;

  const float* x  = (const float*)d_in[0];
  const float* w1 = (const float*)d_in[1];
  const float* w2 = (const float*)d_in[2];
  float* out = (float*)d_out;

  float* A  = (float*)d_ws;                       // [B,P,D] = 3.28 MB
  float* Bm = A + (size_t)B_ * P_ * D_;           // [B,P,D] = 3.28 MB

  dim3 gg((D_ + 15) / 16, P_ / 16, B_);           // (13, 4, 64), 1 wave/block
  gemm_wmma<0><<<gg, 32, 0, stream>>>(x, w1, A);
  gemm_wmma<1><<<gg, 32, 0, stream>>>(x, w2, Bm);

  dim3 go((DD_ / 4 + 255) / 256, B_ * P_, 1);     // (40, 4096)
  outer_sum<<<go, 256, 0, stream>>>(A, Bm, out);
}